// GraphAutoEncoder_17248588660806
// MI455X (gfx1250) — compile-verified
//
#include <hip/hip_runtime.h>

// GCN-style pipeline, reassociated so every dense GEMM precedes its SpMM:
//   Y0 = X@W0^T ; H1 = A@Y0 ; Y1 = relu(H1)@W1^T ; H2 = A@Y1 ;
//   Y2 = relu(H2)@W2^T ; out = A@Y2
// GEMMs use V_WMMA_F32_16X16X4_F32 (fp32 matrix core, no precision loss).
// SpMM uses coalesced b128 row gathers + global_atomic_add_f32 scatters
// (output + gather source are L2-resident: 51.2 MB each vs 192 MB L2).

#define N_NODES 100000

typedef __attribute__((ext_vector_type(2))) float v2f;
typedef __attribute__((ext_vector_type(8))) float v8f;

// ---------------------------------------------------------------------------
// Y[m0:m0+16, d0:d0+16] = act(A[m0:m0+16, :K]) @ W[d0:d0+16, :K]^T
// One wave per 16x16 tile. ISA 32-bit fragment layouts (CDNA5 ISA):
//   A 16x4 : lane l<16 holds (M=l, K=k+0..1); lane l+16 holds (M=l, K=k+2..3)
//   B 4x16 : lane l<16 holds (N=l, K=k+0..1); lane l+16 holds (N=l, K=k+2..3)
//   C 16x16: VGPR r, lane l<16 -> (M=r, N=l); lane l+16 -> (M=r+8, N=l)
// ---------------------------------------------------------------------------
template <int K, bool RELU>
__global__ __launch_bounds__(128) void gemm_wmma_f32(
    const float* __restrict__ A,   // [N, K] row-major
    const float* __restrict__ W,   // [D, K] row-major
    float* __restrict__ Y,         // [N, D] row-major
    int Dtot)
{
  const int lane = threadIdx.x;        // 0..31 (wave32)
  const int l    = lane & 15;
  const int half = lane >> 4;          // 0 or 1
  const int m0   = blockIdx.x * 16;
  const int d0   = (blockIdx.y * blockDim.y + threadIdx.y) * 16;

  const float* arow = A + (size_t)(m0 + l) * K + 2 * half;
  const float* wrow = W + (size_t)(d0 + l) * K + 2 * half;

  v8f acc = {};
#pragma unroll 8
  for (int k = 0; k < K; k += 4) {
    v2f av = *(const v2f*)(arow + k);
    if (RELU) {
      av.x = av.x > 0.f ? av.x : 0.f;
      av.y = av.y > 0.f ? av.y : 0.f;
    }
    v2f bv = *(const v2f*)(wrow + k);
    // 8 args: (neg_a, A, neg_b, B, c_mod, C, reuse_a, reuse_b)
    acc = __builtin_amdgcn_wmma_f32_16x16x4_f32(
        false, av, false, bv, (short)0, acc, false, false);
  }

  float* yout = Y + (size_t)(m0 + 8 * half) * Dtot + d0 + l;
#pragma unroll
  for (int r = 0; r < 8; ++r)
    yout[(size_t)r * Dtot] = acc[r];
}

// ---------------------------------------------------------------------------
// H[row[e], :] += vals[e] * Y[col[e], :]   (D = 128 floats per row)
// One wave per edge: lane does one 16B load (wave covers the whole 512B row)
// and 4 fp32 atomic adds. blockDim = (32, EDGES_PER_BLOCK).
// ---------------------------------------------------------------------------
#define EDGES_PER_BLOCK 8

__global__ __launch_bounds__(32 * EDGES_PER_BLOCK) void spmm_scatter(
    const int* __restrict__ row, const int* __restrict__ col,
    const float* __restrict__ vals, const float* __restrict__ Y,
    float* __restrict__ H, int E)
{
  const int e = blockIdx.x * EDGES_PER_BLOCK + threadIdx.y;
  if (e >= E) return;
  const int   r = row[e];
  const int   c = col[e];
  const float v = vals[e];

  const float4 y = ((const float4*)(Y + (size_t)c * 128))[threadIdx.x];
  float* dst = H + (size_t)r * 128 + threadIdx.x * 4;
  atomicAdd(dst + 0, v * y.x);
  atomicAdd(dst + 1, v * y.y);
  atomicAdd(dst + 2, v * y.z);
  atomicAdd(dst + 3, v * y.w);
}

__global__ void zero_f32x4(float4* __restrict__ p, int n4) {
  int i = blockIdx.x * blockDim.x + threadIdx.x;
  if (i < n4) p[i] = make_float4(0.f, 0.f, 0.f, 0.f);
}

// ---------------------------------------------------------------------------
extern "C" void kernel_launch(void* const* d_in, const int* in_sizes, int n_in,
                              void* d_out, int out_size, void* d_ws, size_t ws_size,
                              hipStream_t stream) {
  const float* X    = (const float*)d_in[0];   // [N, 256]
  const int*   row  = (const int*)  d_in[1];   // [E]
  const int*   col  = (const int*)  d_in[2];   // [E]
  const float* vals = (const float*)d_in[3];   // [E]
  const float* W0   = (const float*)d_in[4];   // [128, 256]
  const float* W1   = (const float*)d_in[5];   // [128, 128]
  const float* W2   = (const float*)d_in[6];   // [128, 128]

  float* out = (float*)d_out;                  // [N, 128]
  float* Y   = (float*)d_ws;                   // [N, 128] scratch (51.2 MB)

  const int N = N_NODES;                       // 100000 (divisible by 16)
  const int E = in_sizes[1];                   // 3,200,000
  const int D = 128;

  const dim3 gblk(32, 4);                      // 4 waves / block
  const dim3 ggrid(N / 16, D / 16 / 4);        // (6250, 2): 8 col tiles total
  const int  zgrid = (N * D / 4 + 255) / 256;
  const int  sgrid = (E + EDGES_PER_BLOCK - 1) / EDGES_PER_BLOCK;
  const dim3 sblk(32, EDGES_PER_BLOCK);

  // Y0 = X @ W0^T                       [K=256]
  gemm_wmma_f32<256, false><<<ggrid, gblk, 0, stream>>>(X, W0, Y, D);
  // H1 = A @ Y0          (into d_out)
  zero_f32x4<<<zgrid, 256, 0, stream>>>((float4*)out, N * D / 4);
  spmm_scatter<<<sgrid, sblk, 0, stream>>>(row, col, vals, Y, out, E);
  // Y1 = relu(H1) @ W1^T                [K=128]
  gemm_wmma_f32<128, true><<<ggrid, gblk, 0, stream>>>(out, W1, Y, D);
  // H2 = A @ Y1          (into d_out)
  zero_f32x4<<<zgrid, 256, 0, stream>>>((float4*)out, N * D / 4);
  spmm_scatter<<<sgrid, sblk, 0, stream>>>(row, col, vals, Y, out, E);
  // Y2 = relu(H2) @ W2^T                [K=128]
  gemm_wmma_f32<128, true><<<ggrid, gblk, 0, stream>>>(out, W2, Y, D);
  // out = A @ Y2         (final)
  zero_f32x4<<<zgrid, 256, 0, stream>>>((float4*)out, N * D / 4);
  spmm_scatter<<<sgrid, sblk, 0, stream>>>(row, col, vals, Y, out, E);
}